// SingleHeadAttention_61727269978697
// MI455X (gfx1250) — compile-verified
//
#include <hip/hip_runtime.h>
#include <hip/hip_bf16.h>
#include <math.h>

// ---------------------------------------------------------------------------
// SingleHeadAttention for MI455X (gfx1250), fp32 path via V_WMMA_F32_16X16X4_F32
// B=4, S=2048, E=1024.
//
// d_out layout: [ out: B*S*E floats ][ aw: B*S*S floats ]
// d_ws  layout: [ q: B*S*E ][ kv: B*S*2E ][ o: B*S*E ]  (128 MB, all fp32)
//
// GEMM: 128x128 tile / 256 threads (8 waves, 2x4 wave grid, 4x2 wmma tiles
// each). K staged in 16-deep slices through ping-pong LDS; K-pairs are
// interleaved in LDS so every WMMA fragment is one aligned ds_load_b64.
// ---------------------------------------------------------------------------

typedef __attribute__((ext_vector_type(2))) float v2f;
typedef __attribute__((ext_vector_type(4))) float v4f;
typedef __attribute__((ext_vector_type(8))) float v8f;

#define TILE_M 128
#define TILE_N 128
#define TILE_K 16

// C[M,N] = A[M,K] * op(B),  op(B) = B[N,K]^T if BT else B[K,N]
template <bool BT>
__global__ __launch_bounds__(256) void gemm_wmma_f32(
    const float* __restrict__ A, const float* __restrict__ Bm,
    float* __restrict__ C, int lda, int ldb, int ldc, int K,
    long strideA, long strideB, long strideC) {
  // K-pair interleaved tiles: As[p*128 + m] = {A[m][2p], A[m][2p+1]}
  __shared__ v2f As[2][(TILE_K / 2) * TILE_M];
  __shared__ v2f Bs[2][(TILE_K / 2) * TILE_N];

  const int tid  = threadIdx.x;
  const int lane = tid & 31;
  const int w    = tid >> 5;
  const int wm   = w & 1;         // wave row (0..1) -> 64 M-rows each
  const int wn   = w >> 1;        // wave col (0..3) -> 32 N-cols each
  const int half = lane >> 4;     // 0: lanes 0-15, 1: lanes 16-31
  const int lm   = lane & 15;

  const float* Ab = A  + (long)blockIdx.z * strideA;
  const float* Bb = Bm + (long)blockIdx.z * strideB;
  float*       Cb = C  + (long)blockIdx.z * strideC;

  const int mtile = blockIdx.y * TILE_M;
  const int ntile = blockIdx.x * TILE_N;

  // staging-register mapping (2 x float4 per thread per matrix)
  const int arow = tid >> 2;            // 0..63 (+64 for i=1)
  const int akq  = (tid & 3) << 2;      // 0,4,8,12
  const int bk   = tid >> 5;            // BN: 0..7 (+8 for i=1)
  const int bn4  = (tid & 31) << 2;     // BN: 0,4,...,124

  v8f acc[4][2];
#pragma unroll
  for (int i = 0; i < 4; ++i)
#pragma unroll
    for (int j = 0; j < 2; ++j) {
      v8f z = {};
      acc[i][j] = z;
    }

  v4f a_reg[2], b_reg[2];

  auto loadA = [&](int k0) {
#pragma unroll
    for (int i = 0; i < 2; ++i)
      a_reg[i] = *(const v4f*)(Ab + (long)(mtile + arow + i * 64) * lda + k0 + akq);
  };
  auto storeA = [&](v2f* dst) {
#pragma unroll
    for (int i = 0; i < 2; ++i) {
      v2f lo, hi;
      lo.x = a_reg[i][0]; lo.y = a_reg[i][1];
      hi.x = a_reg[i][2]; hi.y = a_reg[i][3];
      dst[((akq >> 1) + 0) * TILE_M + arow + i * 64] = lo;
      dst[((akq >> 1) + 1) * TILE_M + arow + i * 64] = hi;
    }
  };
  auto loadB = [&](int k0) {
#pragma unroll
    for (int i = 0; i < 2; ++i) {
      if (BT)
        b_reg[i] = *(const v4f*)(Bb + (long)(ntile + arow + i * 64) * ldb + k0 + akq);
      else
        b_reg[i] = *(const v4f*)(Bb + (long)(k0 + bk + i * 8) * ldb + ntile + bn4);
    }
  };
  auto storeB = [&](v2f* dst) {
#pragma unroll
    for (int i = 0; i < 2; ++i) {
      if (BT) {
        v2f lo, hi;
        lo.x = b_reg[i][0]; lo.y = b_reg[i][1];
        hi.x = b_reg[i][2]; hi.y = b_reg[i][3];
        dst[((akq >> 1) + 0) * TILE_N + arow + i * 64] = lo;
        dst[((akq >> 1) + 1) * TILE_N + arow + i * 64] = hi;
      } else {
        const int k = bk + i * 8;
        float* d = (float*)dst;
#pragma unroll
        for (int j = 0; j < 4; ++j)
          d[((k >> 1) * TILE_N + bn4 + j) * 2 + (k & 1)] = b_reg[i][j];
      }
    }
  };
  auto prefetch = [&](int k0) {
    __builtin_prefetch(Ab + (long)(mtile + arow) * lda + k0 + akq, 0, 0);
    if (BT)
      __builtin_prefetch(Bb + (long)(ntile + arow) * ldb + k0 + akq, 0, 0);
    else
      __builtin_prefetch(Bb + (long)(k0 + bk) * ldb + ntile + bn4, 0, 0);
  };
  auto compute = [&](int cur) {
    const v2f* AsP = As[cur];
    const v2f* BsP = Bs[cur];
#pragma unroll
    for (int kk = 0; kk < TILE_K; kk += 4) {
      const int p = (kk >> 1) + half;  // lanes 0-15: K={kk,kk+1}; 16-31: K={kk+2,kk+3}
      v2f bf[2], af[4];
#pragma unroll
      for (int j = 0; j < 2; ++j)
        bf[j] = BsP[p * TILE_N + wn * 32 + j * 16 + lm];
#pragma unroll
      for (int i = 0; i < 4; ++i)
        af[i] = AsP[p * TILE_M + wm * 64 + i * 16 + lm];
#pragma unroll
      for (int i = 0; i < 4; ++i)
#pragma unroll
        for (int j = 0; j < 2; ++j)
          acc[i][j] = __builtin_amdgcn_wmma_f32_16x16x4_f32(
              false, af[i], false, bf[j], (short)0, acc[i][j], false, false);
    }
  };

  // prologue: tile 0 into buffer 0
  loadA(0);
  loadB(0);
  storeA(As[0]);
  storeB(Bs[0]);
  __syncthreads();

  const int T = K >> 4;
  for (int t = 0; t < T; ++t) {
    const int cur = t & 1;
    if (t + 1 < T) {            // issue next tile's global loads early
      loadA((t + 1) << 4);
      loadB((t + 1) << 4);
    }
    if (t + 2 < T) prefetch((t + 2) << 4);  // pull tile t+2 toward GL2
    compute(cur);               // 32 wmma hide the global-load latency
    if (t + 1 < T) {            // drain staged regs into the other buffer
      storeA(As[1 - cur]);
      storeB(Bs[1 - cur]);
    }
    __syncthreads();
  }

  // store: VGPR r -> row r (lanes 0-15) / row r+8 (lanes 16-31)
#pragma unroll
  for (int i = 0; i < 4; ++i)
#pragma unroll
    for (int j = 0; j < 2; ++j) {
      const int col = ntile + wn * 32 + j * 16 + lm;
#pragma unroll
      for (int r = 0; r < 8; ++r) {
        const int row = mtile + wm * 64 + i * 16 + r + half * 8;
        Cb[(long)row * ldc + col] = acc[i][j][r];
      }
    }
}

// In-place row softmax: aw[r,:] = softmax((aw[r,:] + mask[r,:]) * scale)
// One 256-thread block per row of length 2048 (2 float4 / thread).
__global__ __launch_bounds__(256) void softmax_mask_rows(
    float* __restrict__ aw, const float* __restrict__ mask, float scale) {
  __shared__ float red[256];
  const long r  = blockIdx.x;
  const int tid = threadIdx.x;
  v4f* rowv       = (v4f*)(aw + r * 2048L);
  const v4f* mrwv = (const v4f*)(mask + r * 2048L);

  float x[8];
  float lmax = -INFINITY;
#pragma unroll
  for (int i = 0; i < 2; ++i) {
    const v4f t  = rowv[tid + i * 256];
    const v4f mt = mrwv[tid + i * 256];
#pragma unroll
    for (int j = 0; j < 4; ++j) {
      x[i * 4 + j] = (t[j] + mt[j]) * scale;
      lmax = fmaxf(lmax, x[i * 4 + j]);
    }
  }
  red[tid] = lmax;
  __syncthreads();
  for (int s = 128; s > 0; s >>= 1) {
    if (tid < s) red[tid] = fmaxf(red[tid], red[tid + s]);
    __syncthreads();
  }
  const float m = red[0];
  __syncthreads();

  float lsum = 0.0f;
#pragma unroll
  for (int i = 0; i < 8; ++i) {
    x[i] = __expf(x[i] - m);
    lsum += x[i];
  }
  red[tid] = lsum;
  __syncthreads();
  for (int s = 128; s > 0; s >>= 1) {
    if (tid < s) red[tid] += red[tid + s];
    __syncthreads();
  }
  const float inv = 1.0f / red[0];
#pragma unroll
  for (int i = 0; i < 2; ++i) {
    v4f o;
#pragma unroll
    for (int j = 0; j < 4; ++j) o[j] = x[i * 4 + j] * inv;
    rowv[tid + i * 256] = o;
  }
}

extern "C" void kernel_launch(void* const* d_in, const int* in_sizes, int n_in,
                              void* d_out, int out_size, void* d_ws,
                              size_t ws_size, hipStream_t stream) {
  constexpr int B = 4, S = 2048, E = 1024;
  constexpr long BSE = (long)B * S * E;  // 8M
  const float* input  = (const float*)d_in[0];
  const float* memory = (const float*)d_in[1];
  const float* mask   = (const float*)d_in[2];
  const float* Wq     = (const float*)d_in[3];
  const float* Wkv    = (const float*)d_in[4];
  const float* Wo     = (const float*)d_in[5];

  float* outp = (float*)d_out;  // [B*S*E]
  float* awp  = outp + BSE;     // [B*S*S]

  float* q_ws  = (float*)d_ws;     // [B*S*E]
  float* kv_ws = q_ws + BSE;       // [B*S*2E] (flat)
  float* o_ws  = kv_ws + 2 * BSE;  // [B*S*E]
  // faithful kv.view(2,B,S,E): split flat buffer in half
  float* k_ws = kv_ws;
  float* v_ws = kv_ws + BSE;

  const dim3 blk(256);

  // 1) q = input @ Wq^T
  gemm_wmma_f32<true><<<dim3(E / 128, (B * S) / 128, 1), blk, 0, stream>>>(
      input, Wq, q_ws, E, E, E, E, 0, 0, 0);

  // 2) kv = memory @ Wkv^T
  gemm_wmma_f32<true><<<dim3((2 * E) / 128, (B * S) / 128, 1), blk, 0, stream>>>(
      memory, Wkv, kv_ws, E, E, 2 * E, E, 0, 0, 0);

  // 3) scores[b] = q[b] @ k[b]^T  -> aw region of d_out (raw logits)
  gemm_wmma_f32<true><<<dim3(S / 128, S / 128, B), blk, 0, stream>>>(
      q_ws, k_ws, awp, E, E, S, E, (long)S * E, (long)S * E, (long)S * S);

  // 4) aw = softmax((scores + mask) * sqrt(E)), in place
  softmax_mask_rows<<<dim3(B * S), blk, 0, stream>>>(awp, mask, 32.0f);

  // 5) o[b] = aw[b] @ v[b]
  gemm_wmma_f32<false><<<dim3(E / 128, S / 128, B), blk, 0, stream>>>(
      awp, v_ws, o_ws, S, E, E, S, (long)S * S, (long)S * E, (long)S * E);

  // 6) out = o @ Wo^T
  gemm_wmma_f32<true><<<dim3(E / 128, (B * S) / 128, 1), blk, 0, stream>>>(
      o_ws, Wo, outp, E, E, E, E, 0, 0, 0);
}